// SoftBoundaryDiceLoss_48954037240563
// MI455X (gfx1250) — compile-verified
//
#include <hip/hip_runtime.h>
#include <hip/hip_bf16.h>
#include <math.h>

#define B_    8
#define C_    12
#define NC_   96
#define H_    256
#define W_    256
#define NPIX_ 65536
#define BIGF  1.0e6f
#define TILE_R 8

// stats layout (float slots): [0..95]=dmin [96..191]=dmax [192..287]=inter
// [288..383]=sumP2 [384..479]=sumD2 ; int slots [480..575]=has_fg
#define ST_DMIN  0
#define ST_DMAX  96
#define ST_INTER 192
#define ST_SP2   288
#define ST_SD2   384
#define ST_HASFG 480

typedef __attribute__((ext_vector_type(2))) float v2f;
typedef __attribute__((ext_vector_type(8))) float v8f;
typedef __attribute__((ext_vector_type(4))) unsigned int u32x4;
typedef __attribute__((ext_vector_type(8))) int i32x8;
typedef __attribute__((ext_vector_type(4))) int i32x4;

// ---------------- TDM: 1-D contiguous tile load (nelem f32) into LDS ----------
__device__ __forceinline__ void tdm_load_1d(unsigned lds_byte_off,
                                            unsigned long long gaddr,
                                            int nelem) {
  u32x4 g0;
  g0[0] = 1u;                                            // count=1, user descriptor
  g0[1] = lds_byte_off;                                  // lds_addr (bytes)
  g0[2] = (unsigned)(gaddr & 0xFFFFFFFFull);             // global_addr[31:0]
  g0[3] = (unsigned)((gaddr >> 32) & 0x01FFFFFFull)      // global_addr[56:32]
        | 0x80000000u;                                   // type=2 ("image")
  i32x8 g1;
  g1[0] = (int)(2u << 16);                               // data_size=4B
  g1[1] = (int)(((unsigned)nelem & 0xFFFFu) << 16);      // tensor_dim0 lo16
  g1[2] = (int)((((unsigned)nelem >> 16) & 0xFFFFu)      // tensor_dim0 hi16
        | (1u << 16));                                   // tensor_dim1 = 1
  g1[3] = (int)(((unsigned)nelem & 0xFFFFu) << 16);      // tile_dim0
  g1[4] = 1;                                             // tile_dim1 = 1
  g1[5] = nelem;                                         // tensor_dim0_stride lo32
  g1[6] = 0;
  g1[7] = 0;
  i32x4 z = {0, 0, 0, 0};
#if defined(__clang_major__) && (__clang_major__ >= 23)
  i32x8 z8 = {0, 0, 0, 0, 0, 0, 0, 0};
  __builtin_amdgcn_tensor_load_to_lds(g0, g1, z, z, z8, 0);
#else
  __builtin_amdgcn_tensor_load_to_lds(g0, g1, z, z, 0);
#endif
}

// ---------------- float atomic min/max via CAS ---------------------------------
__device__ __forceinline__ void atomicMinF(float* p, float v) {
  int* ip = (int*)p;
  int old = __float_as_int(*p);
  while (v < __int_as_float(old)) {
    int assumed = old;
    old = atomicCAS(ip, assumed, __float_as_int(v));
    if (old == assumed) break;
  }
}
__device__ __forceinline__ void atomicMaxF(float* p, float v) {
  int* ip = (int*)p;
  int old = __float_as_int(*p);
  while (v > __int_as_float(old)) {
    int assumed = old;
    old = atomicCAS(ip, assumed, __float_as_int(v));
    if (old == assumed) break;
  }
}

// ---------------- K0: init stats ----------------------------------------------
__global__ void k_init(float* st) {
  int i = threadIdx.x;
  if (i < NC_) {
    st[ST_DMIN + i]  =  3.0e38f;
    st[ST_DMAX + i]  = -3.0e38f;
    st[ST_INTER + i] = 0.0f;
    st[ST_SP2 + i]   = 0.0f;
    st[ST_SD2 + i]   = 0.0f;
    ((int*)st)[ST_HASFG + i] = 0;
  }
}

// ---------------- K1a: vertical downward scan (nearest zero at-or-above) -------
// writes gA = (h - last_bg_row) for mask m, gB = same for ~m; sets has_fg
__global__ void k_vert_down(const int* __restrict__ tgt,
                            float* __restrict__ gA, float* __restrict__ gB,
                            float* st) {
  int n = blockIdx.x;           // b*C + c
  int w = threadIdx.x;
  int b = n / C_, c = n - b * C_;
  const int* tcol = tgt + b * NPIX_ + w;
  float* pa = gA + (size_t)n * NPIX_ + w;
  float* pb = gB + (size_t)n * NPIX_ + w;
  float lastZp = -BIGF, lastZn = -BIGF;   // nearest zero row index so far
  int any = 0;
  for (int h = 0; h < H_; ++h) {
    int fg = (tcol[h * W_] == c);
    any |= fg;
    float fh = (float)h;
    if (!fg) lastZp = fh;                 // zeros of m = background
    if (fg)  lastZn = fh;                 // zeros of ~m = foreground
    pa[h * W_] = fh - lastZp;
    pb[h * W_] = fh - lastZn;
  }
  __shared__ int sf;
  if (threadIdx.x == 0) sf = 0;
  __syncthreads();
  if (any) sf = 1;
  __syncthreads();
  if (threadIdx.x == 0 && sf) atomicOr(&((int*)st)[ST_HASFG + n], 1);
}

// ---------------- K1b: vertical upward scan, finish 1D distance, square --------
__global__ void k_vert_up(const int* __restrict__ tgt,
                          float* __restrict__ gA, float* __restrict__ gB) {
  int n = blockIdx.x;
  int w = threadIdx.x;
  int b = n / C_, c = n - b * C_;
  const int* tcol = tgt + b * NPIX_ + w;
  float* pa = gA + (size_t)n * NPIX_ + w;
  float* pb = gB + (size_t)n * NPIX_ + w;
  float nextZp = BIGF, nextZn = BIGF;
  for (int h = H_ - 1; h >= 0; --h) {
    int fg = (tcol[h * W_] == c);
    float fh = (float)h;
    if (!fg) nextZp = fh;
    if (fg)  nextZn = fh;
    float gp = fminf(fminf(pa[h * W_], nextZp - fh), BIGF);
    float gn = fminf(fminf(pb[h * W_], nextZn - fh), BIGF);
    pa[h * W_] = gp * gp;                 // g2 for pos EDT
    pb[h * W_] = gn * gn;                 // g2 for neg EDT
  }
}

// ---------------- K2: row min-plus convolution (exact parabola envelope) -------
// TDM stages an 8x256 f32 tile of g2pos and g2neg into LDS; each thread owns
// output column j; dt = sqrt(d2neg)-sqrt(d2pos), zeroed if !has_fg; per-(b,c)
// min/max via CAS atomics. dt written back over gA.
__global__ void k_rows(float* __restrict__ gA, const float* __restrict__ gB,
                       float* st) {
  __shared__ float lp[TILE_R * W_];
  __shared__ float ln[TILE_R * W_];
  __shared__ float smn[8], smx[8];
  int n    = blockIdx.x >> 5;             // /32 tiles per image
  int tb   = blockIdx.x & 31;
  int row0 = tb * TILE_R;
  const float* srcA = gA + (size_t)n * NPIX_ + (size_t)row0 * W_;
  const float* srcB = gB + (size_t)n * NPIX_ + (size_t)row0 * W_;

  if (threadIdx.x < 32) {                 // wave 0 issues both DMAs
    tdm_load_1d((unsigned)(uintptr_t)&lp[0],
                (unsigned long long)(uintptr_t)srcA, TILE_R * W_);
    tdm_load_1d((unsigned)(uintptr_t)&ln[0],
                (unsigned long long)(uintptr_t)srcB, TILE_R * W_);
    __builtin_amdgcn_s_wait_tensorcnt(0);
  }
  __syncthreads();

  int hf = ((const int*)st)[ST_HASFG + n];
  int j = threadIdx.x;
  float jf = (float)j;
  float dmn = 3.0e38f, dmx = -3.0e38f;
  float* out = gA + (size_t)n * NPIX_ + (size_t)row0 * W_;

  for (int r = 0; r < TILE_R; ++r) {
    const float* rp = lp + r * W_;
    const float* rn = ln + r * W_;
    float dp = 3.0e38f, dn = 3.0e38f;
    float dd = jf;                        // j - k at k=0
#pragma unroll 8
    for (int k = 0; k < W_; ++k) {
      float q = dd * dd;
      dp = fminf(dp, q + rp[k]);
      dn = fminf(dn, q + rn[k]);
      dd -= 1.0f;
    }
    float dt = sqrtf(dn) - sqrtf(dp);
    dt = hf ? dt : 0.0f;
    out[r * W_ + j] = dt;
    dmn = fminf(dmn, dt);
    dmx = fmaxf(dmx, dt);
  }

  // block min/max reduction -> CAS atomics
  for (int m = 16; m > 0; m >>= 1) {
    dmn = fminf(dmn, __shfl_xor(dmn, m, 32));
    dmx = fmaxf(dmx, __shfl_xor(dmx, m, 32));
  }
  int wid = threadIdx.x >> 5, lane = threadIdx.x & 31;
  if (lane == 0) { smn[wid] = dmn; smx[wid] = dmx; }
  __syncthreads();
  if (threadIdx.x == 0) {
    float a = smn[0], bx = smx[0];
    for (int i = 1; i < 8; ++i) { a = fminf(a, smn[i]); bx = fmaxf(bx, smx[i]); }
    atomicMinF(&st[ST_DMIN + n], a);
    atomicMaxF(&st[ST_DMAX + n], bx);
  }
}

// ---------------- K3: softmax + normalize + dice sums (WMMA ones-reduction) ----
// One thread per pixel of one image b. 36 quantities per block reduced with
// chains of V_WMMA_F32_16X16X4_F32 against an all-ones B matrix (exact f32
// row sums), then one f32 atomicAdd per quantity per block.
__global__ void k_reduce(const float* __restrict__ logits,
                         const float* __restrict__ dt, float* st) {
  __shared__ float part[36 * 256];        // 36 KB
  int b   = blockIdx.x >> 8;
  int pix = ((blockIdx.x & 255) << 8) + threadIdx.x;

  float lg[C_];
  float mx = -3.0e38f;
#pragma unroll
  for (int c = 0; c < C_; ++c) {
    lg[c] = logits[((b * C_ + c) << 16) + pix];
    mx = fmaxf(mx, lg[c]);
  }
  float s = 0.0f;
#pragma unroll
  for (int c = 0; c < C_; ++c) { lg[c] = expf(lg[c] - mx); s += lg[c]; }
  float inv = 1.0f / s;

#pragma unroll
  for (int c = 0; c < C_; ++c) {
    int n = b * C_ + c;
    float p  = lg[c] * inv;
    float d0 = st[ST_DMIN + n];
    float d1 = st[ST_DMAX + n];
    float d  = dt[((size_t)n << 16) + pix];
    float dn = (d - d0) / (d1 - d0 + 1e-8f);
    part[(c * 3 + 0) * 256 + threadIdx.x] = p * dn;
    part[(c * 3 + 1) * 256 + threadIdx.x] = p * p;
    part[(c * 3 + 2) * 256 + threadIdx.x] = dn * dn;
  }
  __syncthreads();

  int wid  = threadIdx.x >> 5;
  int lane = threadIdx.x & 31;
  int rsel = lane & 15;                   // A-matrix row (M) for this lane
  int koff = (lane >> 4) << 1;            // lanes 16-31 hold K=2,3
  v2f ones = {1.0f, 1.0f};

  for (int q = wid; q < 36; q += 8) {     // uniform per wave -> EXEC all ones
    const float* src = part + q * 256;
    v8f acc = {0.f, 0.f, 0.f, 0.f, 0.f, 0.f, 0.f, 0.f};
#pragma unroll
    for (int ch = 0; ch < 4; ++ch) {
      v2f A;
      A[0] = src[ch * 64 + rsel * 4 + koff];
      A[1] = src[ch * 64 + rsel * 4 + koff + 1];
      // D[i][j] += sum_k A[i][k] : exact f32 row-sum via ones matmul
      acc = __builtin_amdgcn_wmma_f32_16x16x4_f32(
          false, A, false, ones, (short)0, acc, false, false);
    }
    // col j of D is identical for all j; rows 0-7 live in this lane,
    // rows 8-15 in lane^16 -> one xor-shuffle finishes the 256-way sum.
    float sv = acc[0] + acc[1] + acc[2] + acc[3] +
               acc[4] + acc[5] + acc[6] + acc[7];
    float tot = sv + __shfl_xor(sv, 16, 32);
    if (lane == 0) {
      int c = q / 3, t = q - 3 * c;
      atomicAdd(&st[ST_INTER + t * 96 + b * C_ + c], tot);
    }
  }
}

// ---------------- K4: final scalar ---------------------------------------------
__global__ void k_final(const float* __restrict__ st, float* __restrict__ out) {
  __shared__ float red[128];
  int i = threadIdx.x;
  float v = 0.0f;
  if (i < NC_) {
    float inter = st[ST_INTER + i];
    float sp2   = st[ST_SP2 + i];
    float sd2   = st[ST_SD2 + i];
    float dice  = 2.0f * inter / (sp2 + sd2 + 1e-6f);
    v = 1.0f - dice;
  }
  red[i] = v;
  __syncthreads();
  if (i == 0) {
    float s = 0.0f;
    for (int k = 0; k < NC_; ++k) s += red[k];
    out[0] = s / (float)NC_;
  }
}

extern "C" void kernel_launch(void* const* d_in, const int* in_sizes, int n_in,
                              void* d_out, int out_size, void* d_ws, size_t ws_size,
                              hipStream_t stream) {
  (void)in_sizes; (void)n_in; (void)out_size; (void)ws_size;
  const float* logits  = (const float*)d_in[0];       // [8,12,256,256] f32
  const int*   targets = (const int*)d_in[1];         // [8,256,256] i32
  float* bufA = (float*)d_ws;                         // g2pos -> dt  (25.2 MB)
  float* bufB = bufA + (size_t)NC_ * NPIX_;           // g2neg        (25.2 MB)
  float* st   = bufB + (size_t)NC_ * NPIX_;           // stats        (2.3 KB)

  k_init     <<<1, 96, 0, stream>>>(st);
  k_vert_down<<<NC_, W_, 0, stream>>>(targets, bufA, bufB, st);
  k_vert_up  <<<NC_, W_, 0, stream>>>(targets, bufA, bufB);
  k_rows     <<<NC_ * (H_ / TILE_R), 256, 0, stream>>>(bufA, bufB, st);
  k_reduce   <<<B_ * 256, 256, 0, stream>>>(logits, bufA, st);
  k_final    <<<1, 128, 0, stream>>>(st, (float*)d_out);
}